// CapsuleConv2d_52587579572433
// MI455X (gfx1250) — compile-verified
//
#include <hip/hip_runtime.h>
#include <hip/hip_bf16.h>

typedef float v2f __attribute__((ext_vector_type(2)));
typedef float v4f __attribute__((ext_vector_type(4)));
typedef float v8f __attribute__((ext_vector_type(8)));

// Problem constants (from reference)
#define BDIM 16
#define IC0  4
#define CIN  64
#define HH   56
#define WW   56
#define OO   64
#define WC1  4
#define OC   256               // WC1 * OO
#define KTOT 576               // CIN * 3 * 3, k = ci*9 + r*3 + s (weights contiguous)
#define NTOT (BDIM*IC0*HH*WW)  // 200704 pixels, p = ((b*56+h)*56+w)*4 + i

// Tiling: block 128(M) x 128(N) x 16(K); 8 wave32 as 4(M) x 2(N); wave tile 32x64
#define BM 128
#define BN 128
#define BK 16
#define APITCH  20   // floats/row; mult of 4 (v4f), 20 banks -> 16 rows on distinct banks
#define BPITCH2 288  // floats per k-PAIR row: 128 cols * 2 + 32 pad; 288%64==32 ->
                     //   half-waves (pair q vs q+1) land in disjoint bank halves
#define KTILES (KTOT/BK)  // 36

__global__ __launch_bounds__(256)
void caps_conv_wmma_f32(const float* __restrict__ X,
                        const float* __restrict__ Wg,
                        float* __restrict__ Out)
{
    __shared__ float As[2][BM * APITCH];      // 10240 B each
    __shared__ float Bs[2][(BK/2) * BPITCH2]; //  9216 B each (8 k-pairs)

    const int tid  = threadIdx.x;
    const int lane = tid & 31;
    const int wave = tid >> 5;
    const int half = lane >> 4;   // 0: lanes 0-15, 1: lanes 16-31
    const int lr   = lane & 15;
    const int wm   = wave & 3;    // wave M index (0..3), 32 rows each
    const int wn   = wave >> 2;   // wave N index (0..1), 64 cols each

    const int nBase = blockIdx.x * BN;   // pixel tile
    const int mBase = blockIdx.y * BM;   // oc tile

    v8f acc[2][4];
    #pragma unroll
    for (int mi = 0; mi < 2; ++mi)
        #pragma unroll
        for (int ni = 0; ni < 4; ++ni)
            acc[mi][ni] = {};

    // ---- hoisted per-thread B-gather state (loop-invariant) ----
    // B tile is 16(k) x 128(p): each thread gathers 8 elements, all at the
    // same pixel column (coalesced across lanes), rows krow = e*2 + (tid>>7).
    const int bCol  = tid & 127;
    const int kRow0 = tid >> 7;          // 0 or 1 (uniform per wave)
    const unsigned p  = (unsigned)(nBase + bCol);
    const unsigned pi = p & 3u;
    const unsigned t1 = p >> 2;
    const unsigned pw = t1 % WW;
    const unsigned t2 = t1 / WW;
    const unsigned ph = t2 % HH;
    const unsigned pb = t2 / HH;
    // element index of x[pb, ci=0, ph-1, pw-1, pi]; may be negative at edges
    const long long baseIdx = ((long long)pb * CIN * (HH * WW)
                               + ((long long)ph - 1) * WW
                               + ((long long)pw - 1)) * IC0 + pi;
    // 3-bit validity masks over r and s (halo handling)
    const unsigned hMask = (ph > 0 ? 1u : 0u) | 2u | (ph < HH - 1 ? 4u : 0u);
    const unsigned wMask = (pw > 0 ? 1u : 0u) | 2u | (pw < WW - 1 ? 4u : 0u);

    // register staging for the global->LDS pipeline
    v4f   aReg[2];
    float bReg[8];

    auto loadA = [&](int k0) {
        #pragma unroll
        for (int q = 0; q < 2; ++q) {
            int s   = tid * 2 + q;        // 512 float4 slots
            int row = s >> 2;             // 0..127
            int kq  = (s & 3) * 4;        // 0,4,8,12
            const float* src = Wg + (size_t)(mBase + row) * KTOT + (k0 + kq);
            aReg[q] = *(const v4f*)src;   // weights fully contiguous in k
        }
    };
    auto loadB = [&](int k0) {
        #pragma unroll
        for (int e = 0; e < 8; ++e) {
            const unsigned k  = (unsigned)(k0 + e * 2 + kRow0);
            const unsigned ci = k / 9u;
            const unsigned rs = k - ci * 9u;
            const unsigned r  = rs / 3u;
            const unsigned s  = rs - r * 3u;
            float v = 0.0f;
            if (((hMask >> r) & (wMask >> s)) & 1u) {
                const long long idx = baseIdx
                    + ((long long)ci * (HH * WW) + (long long)r * WW + s) * IC0;
                v = X[idx];
            }
            bReg[e] = v;
        }
    };
    auto stageTiles = [&](int buf) {
        #pragma unroll
        for (int q = 0; q < 2; ++q) {
            int s   = tid * 2 + q;
            int row = s >> 2;
            int kq  = (s & 3) * 4;
            *(v4f*)&As[buf][row * APITCH + kq] = aReg[q];
        }
        // element for k = e*2 + kRow0 lands at pair e, parity kRow0
        #pragma unroll
        for (int e = 0; e < 8; ++e)
            Bs[buf][e * BPITCH2 + bCol * 2 + kRow0] = bReg[e];
    };
    auto compute = [&](int buf) {
        const float* Ab = &As[buf][0];
        const float* Bb = &Bs[buf][0];
        #pragma unroll
        for (int kk = 0; kk < BK; kk += 4) {
            const int kc    = kk + 2 * half;       // K=0,1 (low) / K=2,3 (high)
            const int kpair = (kk >> 1) + half;    // pair row holding {kc, kc+1}
            v2f af[2], bf[4];
            #pragma unroll
            for (int mi = 0; mi < 2; ++mi) {
                int row = wm * 32 + mi * 16 + lr;             // A: M striped, K in-lane
                af[mi] = *(const v2f*)&Ab[row * APITCH + kc]; // ds_load_b64
            }
            #pragma unroll
            for (int ni = 0; ni < 4; ++ni) {
                int col = wn * 64 + ni * 16 + lr;             // B: N across lanes
                bf[ni] = *(const v2f*)&Bb[kpair * BPITCH2 + col * 2]; // ds_load_b64
            }
            #pragma unroll
            for (int mi = 0; mi < 2; ++mi)
                #pragma unroll
                for (int ni = 0; ni < 4; ++ni)
                    acc[mi][ni] = __builtin_amdgcn_wmma_f32_16x16x4_f32(
                        /*neg_a=*/false, af[mi],
                        /*neg_b=*/false, bf[ni],
                        /*c_mod=*/(short)0, acc[mi][ni],
                        /*reuse_a=*/false, /*reuse_b=*/false);
        }
    };

    // ---- pipelined main loop: single barrier per K-step, double-buffered LDS ----
    loadA(0);
    loadB(0);
    int buf = 0;
    for (int kt = 0; kt < KTILES; ++kt) {
        stageTiles(buf);
        __syncthreads();
        if (kt + 1 < KTILES) {
            loadA((kt + 1) * BK);
            loadB((kt + 1) * BK);
        }
        compute(buf);
        buf ^= 1;
    }

    // ---- epilogue: scatter C tiles into [16,64,56,56,4,4] ----
    #pragma unroll
    for (int ni = 0; ni < 4; ++ni) {
        const unsigned cp  = (unsigned)(nBase + wn * 64 + ni * 16 + lr);
        const unsigned ci_ = cp & 3u;          // input-capsule row i
        const unsigned u1  = cp >> 2;
        const unsigned cw  = u1 % WW;
        const unsigned u2  = u1 / WW;
        const unsigned ch  = u2 % HH;
        const unsigned cb  = u2 / HH;
        const size_t pixBase = ((((size_t)cb * OO) * HH + ch) * WW + cw) * (IC0 * WC1)
                               + ci_ * WC1;
        #pragma unroll
        for (int mi = 0; mi < 2; ++mi) {
            const int ocBase = mBase + wm * 32 + mi * 16 + half * 8;
            #pragma unroll
            for (int v = 0; v < 8; ++v) {
                const int oc = ocBase + v;
                const int j  = oc >> 6;        // wc1 column
                const int o  = oc & 63;        // output capsule channel
                Out[pixBase + (size_t)o * (HH * WW * IC0 * WC1) + j] = acc[mi][ni][v];
            }
        }
    }
}

extern "C" void kernel_launch(void* const* d_in, const int* in_sizes, int n_in,
                              void* d_out, int out_size, void* d_ws, size_t ws_size,
                              hipStream_t stream) {
    (void)in_sizes; (void)n_in; (void)d_ws; (void)ws_size; (void)out_size;
    const float* X  = (const float*)d_in[0];   // [16,64,56,56,4,1] fp32
    const float* Wg = (const float*)d_in[1];   // [4,64,64,3,3]    fp32 (== [256,576])
    float* Out = (float*)d_out;                // [16,64,56,56,4,4] fp32

    dim3 grid(NTOT / BN, OC / BM, 1);          // (1568, 2)
    dim3 block(256, 1, 1);                     // 8 wave32
    caps_conv_wmma_f32<<<grid, block, 0, stream>>>(X, Wg, Out);
}